// EigenCAMAttention_70145405878609
// MI455X (gfx1250) — compile-verified
//
#include <hip/hip_runtime.h>
#include <hip/hip_bf16.h>
#include <math.h>

// Problem constants (from reference)
#define B_   64
#define C_   512
#define HW_  1024
#define R_   32

typedef __attribute__((ext_vector_type(2))) float v2f;
typedef __attribute__((ext_vector_type(8))) float v8f;
typedef __attribute__((ext_vector_type(4))) int   v4i;

#define AS1 __attribute__((address_space(1)))
#define AS3 __attribute__((address_space(3)))

// ---- gfx1250 data-mover feature detection (guarded; fallbacks keep us compiling) ----
#if defined(__gfx1250__) && __has_builtin(__builtin_amdgcn_tensor_load_to_lds) && \
    __has_builtin(__builtin_amdgcn_s_wait_tensorcnt)
#define HAVE_TDM 1
typedef __attribute__((ext_vector_type(4))) unsigned int u32x4;
typedef __attribute__((ext_vector_type(4))) int i32x4;
typedef __attribute__((ext_vector_type(8))) int i32x8;
#endif

#if defined(__gfx1250__) && __has_builtin(__builtin_amdgcn_global_load_async_to_lds_b128)
#define HAVE_ASYNC_LDS 1
#endif

__device__ __forceinline__ void async_wait_all() {
#if defined(__gfx1250__)
#if __has_builtin(__builtin_amdgcn_s_wait_asynccnt)
  __builtin_amdgcn_s_wait_asynccnt(0);
#else
  asm volatile("s_wait_asynccnt 0x0" ::: "memory");
#endif
#endif
}

// ---------------- wave32 reductions ----------------
__device__ __forceinline__ float wave_sum(float s) {
#pragma unroll
  for (int off = 16; off > 0; off >>= 1) s += __shfl_xor(s, off, 32);
  return s;
}
__device__ __forceinline__ float wave_min(float s) {
#pragma unroll
  for (int off = 16; off > 0; off >>= 1) s = fminf(s, __shfl_xor(s, off, 32));
  return s;
}
__device__ __forceinline__ float wave_max(float s) {
#pragma unroll
  for (int off = 16; off > 0; off >>= 1) s = fmaxf(s, __shfl_xor(s, off, 32));
  return s;
}
__device__ __forceinline__ float sigmoidf(float x) { return 1.0f / (1.0f + expf(-x)); }

// ---------------- K1: pooled[b,c] = mean_hw x ----------------
__global__ void k1_pool(const float* __restrict__ x, float* __restrict__ pooled) {
  const int wid = threadIdx.x >> 5, lane = threadIdx.x & 31;
  const int row = blockIdx.x * 8 + wid;            // [0, B_*C_)
  const float* p = x + (size_t)row * HW_;
  float s = 0.0f;
  for (int i = lane; i < HW_; i += 32) s += p[i];
  s = wave_sum(s);
  if (lane == 0) pooled[row] = s * (1.0f / (float)HW_);
}

// ---------------- K2: channel attention, one block per batch ----------------
__global__ void k2_chatt(const float* __restrict__ pooled,
                         const float* __restrict__ w_fc1, const float* __restrict__ b_fc1,
                         const float* __restrict__ w_fc2, const float* __restrict__ b_fc2,
                         float* __restrict__ ca) {
  __shared__ float sp[C_];
  __shared__ float sca1[R_];
  const int b = blockIdx.x;
  for (int i = threadIdx.x; i < C_; i += 256) sp[i] = pooled[b * C_ + i];
  __syncthreads();
  const int wid = threadIdx.x >> 5, lane = threadIdx.x & 31;
  for (int r = wid; r < R_; r += 8) {
    float s = 0.0f;
    for (int c = lane; c < C_; c += 32) s += sp[c] * w_fc1[r * C_ + c];
    s = wave_sum(s);
    if (lane == 0) sca1[r] = fmaxf(s + b_fc1[r], 0.0f);
  }
  __syncthreads();
  for (int c = threadIdx.x; c < C_; c += 256) {
    float s = b_fc2[c];
#pragma unroll
    for (int r = 0; r < R_; ++r) s += sca1[r] * w_fc2[c * R_ + r];
    ca[b * C_ + c] = sigmoidf(s);
  }
}

// ---------------- K3: feat[b] = w_eig @ x[b] + b_eig  (WMMA f32 16x16x4) ----------------
// grid (HW_/64, B_), block 256 = 8 waves; wave = (mt in 0..1) x (nt in 0..3) -> 16x16 tile
__global__ void k3_feat_wmma(const float* __restrict__ x,
                             const float* __restrict__ w_eig, const float* __restrict__ b_eig,
                             float* __restrict__ feat) {
  __shared__ float sA[R_ * C_];                    // 64 KB: full w_eig in LDS
#ifdef HAVE_ASYNC_LDS
  // gfx1250 async global->LDS copy (ASYNCcnt path), 16B per lane per issue
  for (int i = threadIdx.x * 4; i < R_ * C_; i += 256 * 4)
    __builtin_amdgcn_global_load_async_to_lds_b128(
        (AS1 v4i*)(w_eig + i), (AS3 v4i*)(sA + i), 0, 0);
  async_wait_all();
#else
  {
    const float4* src = (const float4*)w_eig;
    float4* dst = (float4*)sA;
    for (int i = threadIdx.x; i < (R_ * C_) / 4; i += 256) dst[i] = src[i];
  }
#endif
  __syncthreads();

  const int b = blockIdx.y;
  const int wid = threadIdx.x >> 5, lane = threadIdx.x & 31;
  const int mt = wid & 1, nt = wid >> 1;
  const int n0 = blockIdx.x * 64 + nt * 16;
  const int lg = lane >> 4;                        // lane group 0/1 -> K offset 0/2
  const int ln = lane & 15;

  const float* Arow  = sA + (size_t)(mt * 16 + ln) * C_;
  const float* Bbase = x + (size_t)b * C_ * HW_ + n0 + ln;

  v8f acc = {};
  for (int k = 0; k < C_; k += 4) {
    if ((k & 15) == 0)                             // stream-ahead hint -> global_prefetch_b8
      __builtin_prefetch(Bbase + (size_t)(k + 16) * HW_, 0, 0);
    v2f a, bb;
    a.x  = Arow[k + lg * 2 + 0];
    a.y  = Arow[k + lg * 2 + 1];
    bb.x = Bbase[(size_t)(k + lg * 2 + 0) * HW_];
    bb.y = Bbase[(size_t)(k + lg * 2 + 1) * HW_];
    acc = __builtin_amdgcn_wmma_f32_16x16x4_f32(false, a, false, bb, (short)0, acc,
                                                false, false);
  }
  const int Nc = n0 + ln;
#pragma unroll
  for (int j = 0; j < 8; ++j) {
    const int M = mt * 16 + j + lg * 8;
    feat[((size_t)b * R_ + M) * HW_ + Nc] = acc[j] + b_eig[M];
  }
}

// ---------------- K3b: spatial_att[b,n] = sigmoid(sum_c x*w_sp + b_sp) ----------------
__global__ void k3b_spatial(const float* __restrict__ x, const float* __restrict__ w_sp,
                            const float* __restrict__ b_sp, float* __restrict__ spo) {
  const int idx = blockIdx.x * 256 + threadIdx.x;  // b*HW + n
  const int b = idx >> 10, n = idx & (HW_ - 1);
  const float* p = x + (size_t)b * C_ * HW_ + n;
  float s = b_sp[0];
  for (int c = 0; c < C_; ++c) s += p[(size_t)c * HW_] * w_sp[c];
  spo[idx] = sigmoidf(s);
}

// ---------------- K4: per-batch cov (WMMA) + power iteration + eigen attention ------
// one block (256 thr) per batch; dynamic LDS holds feat_c (128KB) + cov + att + scratch
__global__ void k4_eigen(const float* __restrict__ feat, const float* __restrict__ v0,
                         const float* __restrict__ spo, float* __restrict__ att_all) {
  extern __shared__ float smem[];
  float* fc     = smem;                 // 32768 : feat_c, 32 rows x 1024
  float* cov    = fc + R_ * HW_;        // 1024  : 32x32
  float* att    = cov + R_ * R_;        // 1024
  float* means  = att + HW_;            // 32
  float* vvec   = means + R_;           // 32
  float* redmin = vvec + R_;            // 8
  float* redmax = redmin + 8;           // 8
  float* scal   = redmax + 8;           // 2

  const int b = blockIdx.x;
  const int wid = threadIdx.x >> 5, lane = threadIdx.x & 31;

  // ---- stage feat[b] (32x1024 f32 = 128KB contiguous) into LDS ----
#if defined(HAVE_TDM)
  // Tensor Data Mover: one 2D descriptor, issued by wave 0; TENSORcnt path.
  if (wid == 0) {
    const unsigned lds_base = (unsigned)(uintptr_t)(AS3 void*)fc;
    const unsigned long long ga =
        (unsigned long long)(uintptr_t)(feat + (size_t)b * R_ * HW_);
    // D# group0: count=1 | lds_addr | global_addr[56:0] | type=2
    u32x4 g0;
    g0[0] = 1u;                                        // count=1, user mode
    g0[1] = lds_base;                                  // lds_addr (bytes)
    g0[2] = (unsigned)(ga & 0xFFFFFFFFu);              // global_addr[31:0]
    g0[3] = (unsigned)((ga >> 32) & 0x01FFFFFFu) | (2u << 30);  // [56:32] | type=2
    // D# group1: data_size=4B, one row of 32768 elements (tile == tensor)
    i32x8 g1;
    g1[0] = (int)(2u << 16);                           // data_size=2 (4 bytes)
    g1[1] = (int)(32768u << 16);                       // tensor_dim0[15:0] -> bits[63:48]
    g1[2] = (int)((32768u >> 16) | (1u << 16));        // tensor_dim0[31:16] | tensor_dim1=1
    g1[3] = (int)(32768u << 16);                       // tile_dim0 = 32768
    g1[4] = 1;                                         // tile_dim1 = 1, tile_dim2 = 0
    g1[5] = 32768;                                     // tensor_dim0_stride[31:0]
    g1[6] = 0;                                         // stride hi / dim1 stride lo
    g1[7] = 0;
    i32x4 g2 = {0, 0, 0, 0};
    i32x4 g3 = {0, 0, 0, 0};
    i32x8 g4 = {0, 0, 0, 0, 0, 0, 0, 0};               // 6-arg form (clang-23 toolchain)
    __builtin_amdgcn_tensor_load_to_lds(g0, g1, g2, g3, g4, 0);
    __builtin_amdgcn_s_wait_tensorcnt(0);
  }
#elif defined(HAVE_ASYNC_LDS)
  for (int i = threadIdx.x * 4; i < R_ * HW_; i += 256 * 4)
    __builtin_amdgcn_global_load_async_to_lds_b128(
        (AS1 v4i*)(feat + (size_t)b * R_ * HW_ + i), (AS3 v4i*)(fc + i), 0, 0);
  async_wait_all();
#else
  {
    const float4* src = (const float4*)(feat + (size_t)b * R_ * HW_);
    float4* dst = (float4*)fc;
    for (int i = threadIdx.x; i < (R_ * HW_) / 4; i += 256) dst[i] = src[i];
  }
#endif
  __syncthreads();

  // row means, then center
  for (int r = wid; r < R_; r += 8) {
    float s = 0.0f;
    for (int i = lane; i < HW_; i += 32) s += fc[r * HW_ + i];
    s = wave_sum(s);
    if (lane == 0) means[r] = s * (1.0f / (float)HW_);
  }
  __syncthreads();
  for (int i = threadIdx.x; i < R_ * HW_; i += 256) fc[i] -= means[i >> 10];
  __syncthreads();

  // cov = fc @ fc^T / 1023 + 1e-5*I   via WMMA f32, waves 0..3 each own a 16x16 tile
  if (wid < 4) {
    const int mt = wid & 1, nt = wid >> 1;
    const int lg = lane >> 4, ln = lane & 15;
    const float* Ar = fc + (size_t)(mt * 16 + ln) * HW_;
    const float* Br = fc + (size_t)(nt * 16 + ln) * HW_;   // B[k,n] = fc[n,k]
    v8f acc = {};
    for (int k = 0; k < HW_; k += 4) {
      v2f a, bb;
      a.x  = Ar[k + lg * 2 + 0];
      a.y  = Ar[k + lg * 2 + 1];
      bb.x = Br[k + lg * 2 + 0];
      bb.y = Br[k + lg * 2 + 1];
      acc = __builtin_amdgcn_wmma_f32_16x16x4_f32(false, a, false, bb, (short)0, acc,
                                                  false, false);
    }
    const int Nc = nt * 16 + ln;
#pragma unroll
    for (int j = 0; j < 8; ++j) {
      const int M = mt * 16 + j + lg * 8;
      cov[M * R_ + Nc] = acc[j] * (1.0f / (float)(HW_ - 1)) + (M == Nc ? 1e-5f : 0.0f);
    }
  }
  __syncthreads();

  // power iteration (wave 0): v = v0/||v0||; 5x { v = cov@v; v /= (||v||+1e-10) }
  if (wid == 0) {
    float v = v0[b * R_ + lane];
    float n2 = wave_sum(v * v);
    v = v / sqrtf(n2);
#pragma unroll
    for (int it = 0; it < 5; ++it) {
      float w = 0.0f;
#pragma unroll
      for (int s2 = 0; s2 < R_; ++s2) w += cov[lane * R_ + s2] * __shfl(v, s2, 32);
      float wn = wave_sum(w * w);
      v = w / (sqrtf(wn) + 1e-10f);
    }
    vvec[lane] = v;
  }
  __syncthreads();

  // att[n] = sum_r v[r] * fc[r,n]
  for (int n = threadIdx.x; n < HW_; n += 256) {
    float s = 0.0f;
#pragma unroll
    for (int r = 0; r < R_; ++r) s += vvec[r] * fc[r * HW_ + n];
    att[n] = s;
  }
  __syncthreads();

  // block min / max
  float lmin = INFINITY, lmax = -INFINITY;
  for (int n = threadIdx.x; n < HW_; n += 256) {
    const float v = att[n];
    lmin = fminf(lmin, v); lmax = fmaxf(lmax, v);
  }
  lmin = wave_min(lmin); lmax = wave_max(lmax);
  if (lane == 0) { redmin[wid] = lmin; redmax[wid] = lmax; }
  __syncthreads();
  if (threadIdx.x == 0) {
    float m0 = redmin[0], m1 = redmax[0];
    for (int w = 1; w < 8; ++w) { m0 = fminf(m0, redmin[w]); m1 = fmaxf(m1, redmax[w]); }
    scal[0] = m0; scal[1] = m1;
  }
  __syncthreads();

  const float amin = scal[0];
  const float denom = (scal[1] - scal[0]) + 1e-10f;   // max(att - amin) + eps
  for (int n = threadIdx.x; n < HW_; n += 256) {
    const float ea = sigmoidf((att[n] - amin) / denom);
    att_all[b * HW_ + n] = ea * spo[b * HW_ + n];
  }
}

// ---------------- K5: out = x * (1 + ca[b,c] * att_all[b,n]) ----------------
__global__ void k5_final(const float* __restrict__ x, const float* __restrict__ ca,
                         const float* __restrict__ att_all, float* __restrict__ out) {
  const size_t i4 = (size_t)blockIdx.x * 256 + threadIdx.x;  // float4 index
  const size_t i = i4 * 4;
  const int b = (int)(i >> 19);                              // C_*HW_ = 2^19
  const int c = (int)((i >> 10) & (C_ - 1));
  const int n = (int)(i & (HW_ - 1));
  const float a = ca[b * C_ + c];
  const float4 xv = ((const float4*)x)[i4];
  const float4 t  = ((const float4*)att_all)[(b * HW_ + n) >> 2];
  float4 o;
  o.x = xv.x * (1.0f + a * t.x);
  o.y = xv.y * (1.0f + a * t.y);
  o.z = xv.z * (1.0f + a * t.z);
  o.w = xv.w * (1.0f + a * t.w);
  ((float4*)out)[i4] = o;
}

extern "C" void kernel_launch(void* const* d_in, const int* in_sizes, int n_in,
                              void* d_out, int out_size, void* d_ws, size_t ws_size,
                              hipStream_t stream) {
  const float* x     = (const float*)d_in[0];
  const float* w_fc1 = (const float*)d_in[1];
  const float* b_fc1 = (const float*)d_in[2];
  const float* w_fc2 = (const float*)d_in[3];
  const float* b_fc2 = (const float*)d_in[4];
  const float* w_eig = (const float*)d_in[5];
  const float* b_eig = (const float*)d_in[6];
  const float* w_sp  = (const float*)d_in[7];
  const float* b_sp  = (const float*)d_in[8];
  const float* v0    = (const float*)d_in[9];
  float* out = (float*)d_out;

  // workspace layout (floats)
  float* ws      = (float*)d_ws;
  float* pooled  = ws;                          // B_*C_   = 32768
  float* ca      = pooled + B_ * C_;            // B_*C_   = 32768
  float* spo     = ca + B_ * C_;                // B_*HW_  = 65536
  float* att_all = spo + B_ * HW_;              // B_*HW_  = 65536
  // feat (B_*R_*HW_ = 8MB) parks in d_out; K5 rewrites all of d_out afterwards
  float* feat = out;

  k1_pool<<<(B_ * C_) / 8, 256, 0, stream>>>(x, pooled);
  k2_chatt<<<B_, 256, 0, stream>>>(pooled, w_fc1, b_fc1, w_fc2, b_fc2, ca);
  k3_feat_wmma<<<dim3(HW_ / 64, B_), 256, 0, stream>>>(x, w_eig, b_eig, feat);
  k3b_spatial<<<(B_ * HW_) / 256, 256, 0, stream>>>(x, w_sp, b_sp, spo);
  const size_t k4_lds = (size_t)(R_ * HW_ + R_ * R_ + HW_ + R_ + R_ + 8 + 8 + 2) * sizeof(float);
  k4_eigen<<<B_, 256, k4_lds, stream>>>(feat, v0, spo, att_all);
  k5_final<<<(B_ * C_ * HW_) / (256 * 4), 256, 0, stream>>>(x, ca, att_all, out);
}